// WindowAttention_86517821213047
// MI455X (gfx1250) — compile-verified
//
#include <hip/hip_runtime.h>

typedef __attribute__((ext_vector_type(2))) float v2f;
typedef __attribute__((ext_vector_type(8))) float v8f;

#define DIMC   192
#define HEADS  6
#define HD     32
#define NT     49      // tokens per 7x7 window
#define NTP    64      // padded tokens
#define DISPC  3
#define IMG    56
#define NWH    8
#define NWW    8
#define XS     196     // LDS stride: x tile / attn-out tile (196 % 64 == 4)
#define QS     580     // LDS stride: qkv tile (580 % 64 == 4)
#define DSS    68      // LDS stride: dots tile (68 % 64 == 4)
#define SCALEF 0.17677669529663687f   // 32^-0.5

// D = A(16x4) * B(4x16) + C, fp32 WMMA (CDNA5)
__device__ __forceinline__ v8f wmma4(v2f a, v2f b, v8f c) {
    return __builtin_amdgcn_wmma_f32_16x16x4_f32(
        /*neg_a=*/false, a, /*neg_b=*/false, b,
        /*c_mod=*/(short)0, c, /*reuse_a=*/false, /*reuse_b=*/false);
}

__device__ __forceinline__ v2f ld2(const float* p) { return *(const v2f*)p; }

__global__ __launch_bounds__(256)
void swin_window_attn_kernel(const float* __restrict__ x,
                             const float* __restrict__ w_qkv,
                             const float* __restrict__ w_out,
                             const float* __restrict__ b_out,
                             const float* __restrict__ pe,
                             float* __restrict__ out)
{
    extern __shared__ float smem[];
    float* s_pe  = smem;                 // 169 (pad 172): 13x13 pos embedding
    float* s_x   = smem + 172;           // 64 x XS : x tile, later attn-out tile
    float* s_qkv = s_x   + NTP * XS;     // 64 x QS : q|k|v (576 cols)
    float* s_dot = s_qkv + NTP * QS;     // 64 x DSS: dots / attn probs

    const int tid  = threadIdx.x;
    const int lane = tid & 31;
    const int wave = tid >> 5;
    const int half = lane >> 4;
    const int lm   = lane & 15;

    const int blk = blockIdx.x;
    const int b   = blk >> 6;
    const int wy  = (blk >> 3) & 7;
    const int wx  = blk & 7;

    // ---------- Phase 0: load bias table + rolled x tile into LDS ----------
    if (tid < 169) s_pe[tid] = pe[tid];
    for (int idx = tid; idx < NTP * 48; idx += 256) {
        const int i  = idx / 48;        // token row
        const int dv = idx % 48;        // float4 index along DIM
        float4 v = make_float4(0.f, 0.f, 0.f, 0.f);
        if (i < NT) {
            const int r = i / 7, c = i % 7;
            int gh = wy * 7 + r + DISPC; if (gh >= IMG) gh -= IMG;
            int gw = wx * 7 + c + DISPC; if (gw >= IMG) gw -= IMG;
            v = *(const float4*)(x + (((size_t)b * IMG + gh) * IMG + gw) * DIMC + dv * 4);
        }
        *(float4*)(s_x + i * XS + dv * 4) = v;
    }
    __syncthreads();

    // ---------- Phase 1: QKV GEMM  (64x192) @ (192x576) -> LDS ----------
    for (int nt = wave; nt < 36; nt += 8) {
        const int nb = nt * 16;
        v8f acc[4] = {};
        for (int k0 = 0; k0 < DIMC; k0 += 4) {
            const int kk = k0 + 2 * half;
            v2f bf;
            bf.x = w_qkv[(size_t)kk       * 576 + nb + lm];
            bf.y = w_qkv[(size_t)(kk + 1) * 576 + nb + lm];
            #pragma unroll
            for (int mt = 0; mt < 4; ++mt) {
                v2f af = ld2(s_x + (mt * 16 + lm) * XS + kk);
                acc[mt] = wmma4(af, bf, acc[mt]);
            }
        }
        #pragma unroll
        for (int mt = 0; mt < 4; ++mt)
            #pragma unroll
            for (int r = 0; r < 8; ++r)
                s_qkv[(mt * 16 + r + 8 * half) * QS + nb + lm] = acc[mt][r];
    }
    __syncthreads();

    // ---------- Phase 2: attention per head ----------
    const bool mask_ul = (wy == NWH - 1);
    const bool mask_lr = (wx == NWW - 1);

    for (int h = 0; h < HEADS; ++h) {
        const int qc = h * HD;
        const int kc = DIMC + h * HD;
        const int vc = 2 * DIMC + h * HD;

        // dots = q @ k^T, fused scale + bias + shift masks
        for (int t = wave; t < 16; t += 8) {
            const int mt = t >> 2, ntl = t & 3;
            v8f acc = {};
            for (int k0 = 0; k0 < HD; k0 += 4) {
                const int kk = k0 + 2 * half;
                v2f a  = ld2(s_qkv + (mt  * 16 + lm) * QS + qc + kk);
                v2f bb = ld2(s_qkv + (ntl * 16 + lm) * QS + kc + kk); // K^T frag
                acc = wmma4(a, bb, acc);
            }
            #pragma unroll
            for (int r = 0; r < 8; ++r) {
                const int i = mt * 16 + r + 8 * half;
                const int j = ntl * 16 + lm;
                float val = 0.f;
                if (i < NT && j < NT) {
                    const int xi = i / 7, yi = i % 7, xj = j / 7, yj = j % 7;
                    val = acc[r] * SCALEF + s_pe[(xj - xi + 6) * 13 + (yj - yi + 6)];
                    if (mask_ul && ((i >= 28) != (j >= 28))) val = -1e30f;
                    if (mask_lr && ((yi >= 4) != (yj >= 4))) val = -1e30f;
                }
                s_dot[i * DSS + j] = val;
            }
        }
        __syncthreads();

        // row softmax: 4 threads per row (all 256 threads busy),
        // width-4 shuffle reductions for max and sum
        {
            const int row = tid >> 2;     // 0..63
            const int q   = tid & 3;      // column quarter
            float* rp = s_dot + row * DSS;
            if (row < NT) {
                float pm = -1e30f;
                #pragma unroll
                for (int jj = 0; jj < 16; ++jj) {
                    const int j = q * 16 + jj;
                    if (j < NT) pm = fmaxf(pm, rp[j]);
                }
                pm = fmaxf(pm, __shfl_xor(pm, 1, 4));
                pm = fmaxf(pm, __shfl_xor(pm, 2, 4));
                float ev[16];
                float ps = 0.f;
                #pragma unroll
                for (int jj = 0; jj < 16; ++jj) {
                    const int j = q * 16 + jj;
                    const float e = (j < NT) ? __expf(rp[j] - pm) : 0.f;
                    ev[jj] = e; ps += e;
                }
                ps += __shfl_xor(ps, 1, 4);
                ps += __shfl_xor(ps, 2, 4);
                const float inv = 1.f / ps;
                #pragma unroll
                for (int jj = 0; jj < 16; ++jj)
                    rp[q * 16 + jj] = ev[jj] * inv;
            } else {
                #pragma unroll
                for (int jj = 0; jj < 16; ++jj)
                    rp[q * 16 + jj] = 0.f;
            }
        }
        __syncthreads();

        // attn(64x64) @ v(64x32) -> ao columns [h*32, h*32+32)  (reuse s_x)
        {
            const int mt = wave >> 1, ntl = wave & 1;
            v8f acc = {};
            for (int k0 = 0; k0 < NTP; k0 += 4) {
                const int kk = k0 + 2 * half;
                v2f a = ld2(s_dot + (mt * 16 + lm) * DSS + kk);
                v2f bb;
                bb.x = s_qkv[kk       * QS + vc + ntl * 16 + lm];
                bb.y = s_qkv[(kk + 1) * QS + vc + ntl * 16 + lm];
                acc = wmma4(a, bb, acc);
            }
            #pragma unroll
            for (int r = 0; r < 8; ++r)
                s_x[(mt * 16 + r + 8 * half) * XS + h * HD + ntl * 16 + lm] = acc[r];
        }
        __syncthreads();
    }

    // ---------- Phase 3: out-proj (64x192)@(192x192) + b_out, rolled store ----
    // 24 duo-tiles (2 M-tiles x 1 N-tile) -> exactly 3 per wave (balanced)
    for (int d = wave; d < 24; d += 8) {
        const int mh = d / 12;            // which pair of M-tiles
        const int nt = d % 12;
        const int nb = nt * 16;
        v8f acc[2] = {};
        for (int k0 = 0; k0 < DIMC; k0 += 4) {
            const int kk = k0 + 2 * half;
            v2f bf;
            bf.x = w_out[(size_t)kk       * DIMC + nb + lm];
            bf.y = w_out[(size_t)(kk + 1) * DIMC + nb + lm];
            #pragma unroll
            for (int mi = 0; mi < 2; ++mi) {
                v2f af = ld2(s_x + ((mh * 2 + mi) * 16 + lm) * XS + kk);
                acc[mi] = wmma4(af, bf, acc[mi]);
            }
        }
        const float bo = b_out[nb + lm];
        #pragma unroll
        for (int mi = 0; mi < 2; ++mi)
            #pragma unroll
            for (int r = 0; r < 8; ++r) {
                const int i = (mh * 2 + mi) * 16 + r + 8 * half;
                if (i < NT) {
                    const int rr = i / 7, cc = i % 7;
                    int gh = wy * 7 + rr + DISPC; if (gh >= IMG) gh -= IMG;
                    int gw = wx * 7 + cc + DISPC; if (gw >= IMG) gw -= IMG;
                    out[(((size_t)b * IMG + gh) * IMG + gw) * DIMC + nb + lm] = acc[mi][r] + bo;
                }
            }
    }
}

extern "C" void kernel_launch(void* const* d_in, const int* in_sizes, int n_in,
                              void* d_out, int out_size, void* d_ws, size_t ws_size,
                              hipStream_t stream) {
    const float* x     = (const float*)d_in[0];
    const float* w_qkv = (const float*)d_in[1];
    const float* w_out = (const float*)d_in[2];
    const float* b_out = (const float*)d_in[3];
    const float* pe    = (const float*)d_in[4];
    float* out = (float*)d_out;

    const size_t shmem = (size_t)(172 + NTP * XS + NTP * QS + NTP * DSS) * sizeof(float);
    hipFuncSetAttribute((const void*)swin_window_attn_kernel,
                        hipFuncAttributeMaxDynamicSharedMemorySize, (int)shmem);

    const int grid = 32 * NWH * NWW;   // one workgroup per (batch, window)
    swin_window_attn_kernel<<<grid, 256, shmem, stream>>>(x, w_qkv, w_out, b_out, pe, out);
}